// PatchShuffle_65712999628933
// MI455X (gfx1250) — compile-verified
//
#include <hip/hip_runtime.h>

// ---------------------------------------------------------------------------
// PatchShuffle for MI455X (gfx1250).
// Pure permutation/memcpy workload (~101 MB traffic -> ~4.3 us at 23.3 TB/s).
// Uses the CDNA5 async global<->LDS DMA path (ASYNCcnt) for the bulk copy:
//   global_load_async_to_lds_b128  ->  s_wait_asynccnt 0  ->
//   global_store_async_from_lds_b128
// fwd/bwd index tensors are computed in closed form (stripe is contiguous,
// so the reference argsort collapses to a branchless formula).
// ---------------------------------------------------------------------------

namespace {
constexpr int kNumRows = 16;
constexpr int kNumCols = 64;
constexpr int kT       = kNumRows * kNumCols;     // 1024
constexpr int kB       = 64;
constexpr int kC       = 768;
constexpr int kStripeW = 48;
constexpr int kKeep    = kNumCols - kStripeW;     // 16 columns survive
constexpr int kVisT    = kNumRows * kKeep;        // 256 visible rows
constexpr int kChunkF4 = kC / 4;                  // 192 float4 per (t,b) chunk
constexpr long long kVisElems = (long long)kVisT * kB * kC;  // 12,582,912 fp32
constexpr int kFwdElems = kT * kB;                // 65,536 int32
constexpr int kWavesPerBlock = 8;                 // 256 threads = 8 wave32
}  // namespace

typedef int v4i __attribute__((ext_vector_type(4)));

#define AS1 __attribute__((address_space(1)))
#define AS3 __attribute__((address_space(3)))

__device__ __forceinline__ AS1 v4i* to_global(const void* p) {
  // generic global address == hardware global address
  return (AS1 v4i*)(unsigned long long)p;
}
__device__ __forceinline__ AS3 v4i* to_lds(const void* p) {
  // generic LDS pointer keeps the LDS byte address in its low 32 bits
  return (AS3 v4i*)(unsigned long long)p;
}

// Column permutation: non-stripe columns first (ascending), then stripe.
__device__ __forceinline__ int perm_col(int j, int s) {
  return (j < s) ? j : ((j < kKeep) ? (j + kStripeW) : (s + (j - kKeep)));
}

// ---------------------------------------------------------------------------
// Kernel 1: fwd / bwd / stripe_bounds (tiny, VALU-only).
// fwd[t*64+b] = r*64 + perm_b[j],  bwd[t*64+b] = r*64 + invperm_b[j]
// ---------------------------------------------------------------------------
__global__ void __launch_bounds__(256)
patchshuffle_indices(const int* __restrict__ start_cols,
                     int* __restrict__ fwd, int* __restrict__ bwd,
                     int* __restrict__ bounds) {
  const int idx = blockIdx.x * 256 + threadIdx.x;   // [0, 65536)
  const int b = idx & (kB - 1);
  const int t = idx >> 6;
  const int r = t >> 6;
  const int j = t & 63;
  const int s = start_cols[b];

  fwd[idx] = (r << 6) + perm_col(j, s);

  // inverse permutation of a column index c = j
  const int inv = (j < s) ? j
                          : ((j < s + kStripeW) ? (kKeep + (j - s))
                                                : (j - kStripeW));
  bwd[idx] = (r << 6) + inv;

  if (idx < kB) {                 // here b == idx
    bounds[idx]      = s;
    bounds[kB + idx] = s + kStripeW;
  }
}

// ---------------------------------------------------------------------------
// Kernel 2: visible = shuffled[:256]  — permuted memcpy of 16384 x 3KB chunks.
// One wave32 per chunk; 6 x B128 per lane each way through an LDS stage using
// the async DMA path (never touches VGPRs for the payload).
// ---------------------------------------------------------------------------
__global__ void __launch_bounds__(256)
patchshuffle_visible_async(const float* __restrict__ patches,
                           const int* __restrict__ start_cols,
                           float* __restrict__ out_visible) {
  __shared__ float4 stage[kWavesPerBlock * kChunkF4];  // 24 KB per block

  const int wave  = threadIdx.x >> 5;
  const int lane  = threadIdx.x & 31;
  const int chunk = blockIdx.x * kWavesPerBlock + wave;  // [0, 16384)

  const int t = chunk >> 6;        // visible row   [0, 256)
  const int b = chunk & (kB - 1);  // batch column
  const int r = t >> 6;            // patch row     [0, 4)
  const int j = t & 63;            // shuffled position within row
  const int s = start_cols[b];
  const int src_t = (r << 6) + perm_col(j, s);

  const float4* src = (const float4*)patches +
                      (long long)(src_t * kB + b) * kChunkF4;
  float4* dst = (float4*)out_visible + (long long)chunk * kChunkF4;
  const float4* lbase = &stage[wave * kChunkF4];

  // memory -> LDS (async DMA, 32 lanes x 16 B x 6 = 3072 B per wave)
#pragma unroll
  for (int k = 0; k < kChunkF4 / 32; ++k) {
    const int e = (k << 5) + lane;
    __builtin_amdgcn_global_load_async_to_lds_b128(to_global(src + e),
                                                   to_lds(lbase + e), 0, 0);
  }
  asm volatile("s_wait_asynccnt 0x0" ::: "memory");

  // LDS -> memory (async DMA); wave-private LDS slice, so no barrier needed
#pragma unroll
  for (int k = 0; k < kChunkF4 / 32; ++k) {
    const int e = (k << 5) + lane;
    __builtin_amdgcn_global_store_async_from_lds_b128(to_global(dst + e),
                                                      to_lds(lbase + e), 0, 0);
  }
  asm volatile("s_wait_asynccnt 0x0" ::: "memory");
}

// ---------------------------------------------------------------------------
// Host launcher. Output layout (flat, 4-byte elems, reference return order):
//   [0, 12582912)              visible  (fp32)
//   [12582912, +65536)         fwd      (int32)
//   [12648448, +65536)         bwd      (int32)
//   [12713984, +128)           stripe_bounds (int32)
// ---------------------------------------------------------------------------
extern "C" void kernel_launch(void* const* d_in, const int* in_sizes, int n_in,
                              void* d_out, int out_size, void* d_ws,
                              size_t ws_size, hipStream_t stream) {
  (void)in_sizes; (void)n_in; (void)out_size; (void)d_ws; (void)ws_size;

  const float* patches    = (const float*)d_in[0];
  const int*   start_cols = (const int*)d_in[1];

  float* out    = (float*)d_out;
  int*   fwd    = (int*)(out + kVisElems);
  int*   bwd    = fwd + kFwdElems;
  int*   bounds = bwd + kFwdElems;

  patchshuffle_indices<<<kFwdElems / 256, 256, 0, stream>>>(start_cols, fwd,
                                                            bwd, bounds);

  patchshuffle_visible_async<<<(kVisT * kB) / kWavesPerBlock, 256, 0, stream>>>(
      patches, start_cols, out);
}